// SafenessLoss_91233695301874
// MI455X (gfx1250) — compile-verified
//
#include <hip/hip_runtime.h>
#include <stdint.h>

#define K_TOT 32
#define KD 31            // number of "others"
#define B_TOT 4096
#define D_TOT 512
#define WAVES_PER_BLOCK 8
#define THREADS (WAVES_PER_BLOCK * 32)

// Issue one 2KB row (512 f32) as 4 async b128 copies, global -> LDS.
// Per the CDNA5 ISA (CDNA5 ISA) the instruction offset is added
// to BOTH the LDS destination address and the global address, so a single
// (lds,global) base pair covers the whole row with offsets 0/512/1024/1536.
__device__ __forceinline__ void async_row(uint32_t lds_byte_addr, uint64_t gaddr) {
    asm volatile(
        "global_load_async_to_lds_b128 %0, %1, off\n\t"
        "global_load_async_to_lds_b128 %0, %1, off offset:512\n\t"
        "global_load_async_to_lds_b128 %0, %1, off offset:1024\n\t"
        "global_load_async_to_lds_b128 %0, %1, off offset:1536"
        :: "v"(lds_byte_addr), "v"(gaddr)
        : "memory");
}

__device__ __forceinline__ float wave_sum(float v) {
    #pragma unroll
    for (int m = 16; m >= 1; m >>= 1) v += __shfl_xor(v, m, 32);
    return v;
}
__device__ __forceinline__ float wave_max(float v) {
    #pragma unroll
    for (int m = 16; m >= 1; m >>= 1) v = fmaxf(v, __shfl_xor(v, m, 32));
    return v;
}

__global__ __launch_bounds__(THREADS)
void safeness_main(const float* __restrict__ emb,
                   const long long* __restrict__ tgt,
                   float* __restrict__ wsL,   // loss * include, per b
                   float* __restrict__ wsI,   // include, per b
                   float* __restrict__ wsC)   // coef * include, per b
{
    __shared__ float buf[WAVES_PER_BLOCK][2][D_TOT];   // 2x 2KB double buffer per wave

    const int lane = threadIdx.x & 31;
    const int wv   = threadIdx.x >> 5;
    const int b    = blockIdx.x * WAVES_PER_BLOCK + wv;

    // Anchor row in registers: lane handles float4 at index c*32 + lane, c=0..3
    const float4* arow = (const float4*)(emb + (size_t)b * D_TOT);
    float4 a0 = arow[lane];
    float4 a1 = arow[32 + lane];
    float4 a2 = arow[64 + lane];
    float4 a3 = arow[96 + lane];

    // LDS byte addresses (generic shared pointer low 32 bits == LDS byte offset)
    const uint32_t lds0 = (uint32_t)(uintptr_t)(&buf[wv][0][0]) + (uint32_t)lane * 16u;
    const uint32_t lds1 = (uint32_t)(uintptr_t)(&buf[wv][1][0]) + (uint32_t)lane * 16u;

    // Prefetch row k=1 into buffer 0
    {
        uint64_t ga = (uint64_t)(uintptr_t)(emb + ((size_t)1 * B_TOT + b) * D_TOT + lane * 4);
        async_row(lds0, ga);
    }

    float my_dist = 0.0f;   // lane j will hold dists[j] (j = 0..30); lane 31 stays 0

    for (int k = 1; k <= KD; ++k) {
        // row k lives in buffer (k-1)&1
        const float4* curv = (const float4*)&buf[wv][(k - 1) & 1][0];

        if (k < KD) {
            uint64_t ga = (uint64_t)(uintptr_t)(emb + ((size_t)(k + 1) * B_TOT + b) * D_TOT + lane * 4);
            async_row((k & 1) ? lds1 : lds0, ga);
            // 8 async ops outstanding; async loads complete in order,
            // so cnt<=4 means the current row's 4 copies have landed in LDS.
            asm volatile("s_wait_asynccnt 0x4" ::: "memory");
        } else {
            asm volatile("s_wait_asynccnt 0x0" ::: "memory");
        }

        float4 o0 = curv[lane];
        float4 o1 = curv[32 + lane];
        float4 o2 = curv[64 + lane];
        float4 o3 = curv[96 + lane];

        float s = 0.0f, d;
        d = a0.x - o0.x; s = fmaf(d, d, s);
        d = a0.y - o0.y; s = fmaf(d, d, s);
        d = a0.z - o0.z; s = fmaf(d, d, s);
        d = a0.w - o0.w; s = fmaf(d, d, s);
        d = a1.x - o1.x; s = fmaf(d, d, s);
        d = a1.y - o1.y; s = fmaf(d, d, s);
        d = a1.z - o1.z; s = fmaf(d, d, s);
        d = a1.w - o1.w; s = fmaf(d, d, s);
        d = a2.x - o2.x; s = fmaf(d, d, s);
        d = a2.y - o2.y; s = fmaf(d, d, s);
        d = a2.z - o2.z; s = fmaf(d, d, s);
        d = a2.w - o2.w; s = fmaf(d, d, s);
        d = a3.x - o3.x; s = fmaf(d, d, s);
        d = a3.y - o3.y; s = fmaf(d, d, s);
        d = a3.z - o3.z; s = fmaf(d, d, s);
        d = a3.w - o3.w; s = fmaf(d, d, s);

        s = wave_sum(s);                 // all lanes now have dists[k-1]
        if (lane == k - 1) my_dist = s;  // lane j keeps dists[j]
    }

    // ---- per-b epilogue, entirely in-wave ----
    const bool valid = (lane < KD);
    const long long t0 = tgt[b];
    const long long tk = valid ? tgt[(size_t)(lane + 1) * B_TOT + b] : (long long)0;

    const bool same = valid && (tk == t0);
    const bool diff = valid && !same;

    const float n_same = (float)__popcll(__ballot(same));
    const float n_diff = (float)KD - n_same;

    const float same_max = wave_max(same ? my_dist : -INFINITY);
    const float same_sum = wave_sum(same ? my_dist : 0.0f);

    const float alpha   = (n_same > 0.0f) ? (same_max + 1.0f) : 1.0f;
    const float shifted = my_dist - alpha;
    const bool  neg     = diff && (shifted < 0.0f);

    const float diff_min_sum = wave_sum(neg ? shifted : 0.0f);
    const int   num_neg      = (int)__popcll(__ballot(neg));

    const float include = ((n_same > 0.0f) || (num_neg > 0)) ? 1.0f : 0.0f;
    const float coef    = n_same / (float)KD;
    const float loss    = (same_sum / (n_same + 1e-6f)
                         - diff_min_sum / (n_diff + 1e-6f)) * (1.0f - coef);

    if (lane == 0) {
        wsL[b] = loss * include;
        wsI[b] = include;
        wsC[b] = coef * include;
    }
}

__global__ __launch_bounds__(256)
void safeness_finalize(const float* __restrict__ wsL,
                       const float* __restrict__ wsI,
                       const float* __restrict__ wsC,
                       float* __restrict__ out)
{
    __shared__ float rl[256], ri[256], rc[256];
    const int t = threadIdx.x;
    float l = 0.0f, i = 0.0f, c = 0.0f;
    for (int b = t; b < B_TOT; b += 256) {
        l += wsL[b]; i += wsI[b]; c += wsC[b];
    }
    rl[t] = l; ri[t] = i; rc[t] = c;
    __syncthreads();
    #pragma unroll
    for (int s = 128; s > 0; s >>= 1) {
        if (t < s) { rl[t] += rl[t + s]; ri[t] += ri[t + s]; rc[t] += rc[t + s]; }
        __syncthreads();
    }
    if (t == 0) {
        const float denom = ri[0];
        out[0] = rl[0] / denom;   // result
        out[1] = rc[0] / denom;   // safety
    }
}

extern "C" void kernel_launch(void* const* d_in, const int* in_sizes, int n_in,
                              void* d_out, int out_size, void* d_ws, size_t ws_size,
                              hipStream_t stream) {
    const float*     emb = (const float*)d_in[0];
    const long long* tgt = (const long long*)d_in[1];
    float* ws  = (float*)d_ws;          // needs 3 * 4096 floats = 48 KB
    float* out = (float*)d_out;

    float* wsL = ws;
    float* wsI = ws + B_TOT;
    float* wsC = ws + 2 * B_TOT;

    safeness_main<<<B_TOT / WAVES_PER_BLOCK, THREADS, 0, stream>>>(emb, tgt, wsL, wsI, wsC);
    safeness_finalize<<<1, 256, 0, stream>>>(wsL, wsI, wsC, out);
}